// MoELayer_25864293057158
// MI455X (gfx1250) — compile-verified
//
#include <hip/hip_runtime.h>
#include <hip/hip_bf16.h>
#include <math.h>

// ---------------- problem constants ----------------
#define D_MODEL 512
#define HIDDEN  2048
#define NEXP    8
#define TOKENS  8192          // B*S = 2*4096
#define TILE_M  32            // tokens per block (2 M-tiles of 16)
#define MAX_TILES (TOKENS / TILE_M)   // 256
#define HCHUNK  256
#define NCHUNKS (HIDDEN / HCHUNK)     // 8
#define XSTRIDE 516           // pad -> conflict-free ds_load_b64 on A tiles
#define HSTRIDE 260

typedef float v2f __attribute__((ext_vector_type(2)));
typedef float v4f __attribute__((ext_vector_type(4)));
typedef float v8f __attribute__((ext_vector_type(8)));

__device__ __forceinline__ v8f vzero8() {
    v8f v = {0.f, 0.f, 0.f, 0.f, 0.f, 0.f, 0.f, 0.f};
    return v;
}

// ---------------- kernel 0: zero output + counters ----------------
__global__ void moe_zero(float* __restrict__ out, int* __restrict__ cnt, int nvec4) {
    int tid = blockIdx.x * blockDim.x + threadIdx.x;
    if (tid < NEXP) cnt[tid] = 0;
    v4f z = {0.f, 0.f, 0.f, 0.f};
    v4f* o = (v4f*)out;
    for (int i = tid; i < nvec4; i += gridDim.x * blockDim.x) o[i] = z;
}

// ---------------- kernel 0b: pre-pack fp32 weights into WMMA-B operand order ----
// src: [Krows_total][N] row-major (expert dim folded into Krows_total; K%4==0 so
// 4-row K-tiles never straddle an expert boundary).
// dst tile (256B) layout: 32 lanes x v2f {src[k+kh][n], src[k+kh+1][n]},
// lane = (kh/2)*16 + n%16  ->  one coalesced global_load_b64 per WMMA B operand.
__global__ void moe_pack(const float* __restrict__ src, float* __restrict__ dst,
                         int ktiles_total, int n16) {
    long total = (long)ktiles_total * n16 * 32;
    long stride = (long)gridDim.x * blockDim.x;
    for (long g = (long)blockIdx.x * blockDim.x + threadIdx.x; g < total; g += stride) {
        int lane = (int)(g & 31);
        long tile = g >> 5;
        int nt  = (int)(tile % n16);
        long kt = tile / n16;
        int N   = n16 * 16;
        int col  = nt * 16 + (lane & 15);
        long krow = kt * 4 + ((lane >> 4) << 1);
        v2f v;
        v.x = src[krow * N + col];
        v.y = src[(krow + 1) * N + col];
        ((v2f*)dst)[tile * 32 + lane] = v;
    }
}

// ---------------- kernel 1: router (softmax + top-2 + compaction) ----------------
__global__ void moe_router(const float* __restrict__ x,
                           const float* __restrict__ ts,
                           const float* __restrict__ gw,   // [D,E]
                           const float* __restrict__ gb,   // [E]
                           int* __restrict__ cnt,
                           int* __restrict__ idx,          // [E][TOKENS]
                           float* __restrict__ wg) {       // [E][TOKENS]
    const int lane = threadIdx.x & 31;
    const int wave = threadIdx.x >> 5;
    const int t = blockIdx.x * (blockDim.x >> 5) + wave;
    if (t >= TOKENS) return;

    float acc[NEXP];
#pragma unroll
    for (int e = 0; e < NEXP; ++e) acc[e] = 0.f;

    const float* xr = x  + (size_t)t * D_MODEL;
    const float* tr = ts + (size_t)t * D_MODEL;
    for (int d = lane; d < D_MODEL; d += 32) {
        float v = xr[d] + tr[d];
        const v4f* g = (const v4f*)(gw + (size_t)d * NEXP);
        v4f g0 = g[0], g1 = g[1];
        acc[0] += v * g0.x; acc[1] += v * g0.y; acc[2] += v * g0.z; acc[3] += v * g0.w;
        acc[4] += v * g1.x; acc[5] += v * g1.y; acc[6] += v * g1.z; acc[7] += v * g1.w;
    }
#pragma unroll
    for (int e = 0; e < NEXP; ++e) {
#pragma unroll
        for (int off = 16; off > 0; off >>= 1)
            acc[e] += __shfl_xor(acc[e], off, 32);
        acc[e] += gb[e];
    }

    if (lane == 0) {
        float m = acc[0];
#pragma unroll
        for (int e = 1; e < NEXP; ++e) m = fmaxf(m, acc[e]);
        float p[NEXP];
        float s = 0.f;
#pragma unroll
        for (int e = 0; e < NEXP; ++e) { p[e] = expf(acc[e] - m); s += p[e]; }
        float inv = 1.f / s;
#pragma unroll
        for (int e = 0; e < NEXP; ++e) p[e] *= inv;

        // top-2, first-index wins ties (matches jax.lax.top_k)
        int i1 = 0;
#pragma unroll
        for (int e = 1; e < NEXP; ++e) if (p[e] > p[i1]) i1 = e;
        int i2 = (i1 == 0) ? 1 : 0;
#pragma unroll
        for (int e = 0; e < NEXP; ++e) if (e != i1 && p[e] > p[i2]) i2 = e;

        int pos1 = atomicAdd(&cnt[i1], 1);
        idx[i1 * TOKENS + pos1] = t;  wg[i1 * TOKENS + pos1] = p[i1];
        int pos2 = atomicAdd(&cnt[i2], 1);
        idx[i2 * TOKENS + pos2] = t;  wg[i2 * TOKENS + pos2] = p[i2];
    }
}

// ---------------- kernel 2: gathered expert MLP via fp32 WMMA ----------------
// 8 waves: quad q = wave&3 owns a column slice; mg = wave>>2 owns an M-tile.
// Waves (q,0) and (q,1) issue identical B addresses -> WGP$ reuse halves L2 traffic.
template <bool PACKED>
__global__ __launch_bounds__(256)
void moe_expert(const float* __restrict__ x,
                const float* __restrict__ w1,   // [E][D][H]
                const float* __restrict__ b1,   // [E][H]
                const float* __restrict__ w2,   // [E][H][D]
                const float* __restrict__ b2,   // [E][D]
                const float* __restrict__ pk1,  // packed W1 (B-operand order)
                const float* __restrict__ pk2,  // packed W2
                const int*   __restrict__ cnt,
                const int*   __restrict__ idx,
                const float* __restrict__ wg,
                float* __restrict__ out) {
    const int e    = blockIdx.y;
    const int tile = blockIdx.x;
    const int count = cnt[e];
    if (tile * TILE_M >= count) return;

    __shared__ float Xs[TILE_M * XSTRIDE];   // gathered 32 x 512 tokens
    __shared__ float Hs[TILE_M * HSTRIDE];   // 32 x 256 gelu(h) chunk
    __shared__ int   toks[TILE_M];
    __shared__ float wgt[TILE_M];

    const int tid  = threadIdx.x;
    const int lane = tid & 31;
    const int wv   = tid >> 5;               // 8 waves
    const int q    = wv & 3;                 // column quad
    const int mg   = wv >> 2;                // M-tile group (0/1)
    const int khalf = (lane >= 16) ? 2 : 0;  // K split for fp32 A/B tiles
    const int mrow  = (lane & 15) + mg * 16; // A-row in LDS
    const int ncol  = lane & 15;             // B col within N-tile

    if (tid < TILE_M) {
        int j = tile * TILE_M + tid;
        if (j < count) { toks[tid] = idx[e * TOKENS + j]; wgt[tid] = wg[e * TOKENS + j]; }
        else           { toks[tid] = 0;                   wgt[tid] = 0.f; }
    }
    __syncthreads();

    // gather X rows: 2 rows per pass, 128 float4 per row
    {
        const int r_off = tid >> 7;      // 0/1
        const int c     = tid & 127;
        for (int r2 = 0; r2 < TILE_M; r2 += 2) {
            int r = r2 + r_off;
            const v4f* src = (const v4f*)(x + (size_t)toks[r] * D_MODEL);
            ((v4f*)(Xs + r * XSTRIDE))[c] = src[c];
        }
    }
    __syncthreads();

    v8f yacc[8];
#pragma unroll
    for (int i = 0; i < 8; ++i) yacc[i] = vzero8();

    const float* w1e = w1 + (size_t)e * D_MODEL * HIDDEN;
    const float* w2e = w2 + (size_t)e * HIDDEN * D_MODEL;

    for (int hc = 0; hc < NCHUNKS; ++hc) {
        // ---- GEMM1: H[16 x 64] per wave = X[16x512] @ W1[512, cols] ----
        v8f hacc[4];
#pragma unroll
        for (int i = 0; i < 4; ++i) hacc[i] = vzero8();

        const int colbase = hc * HCHUNK + q * 64;          // first of 4 N-tiles
        for (int kt = 0; kt < D_MODEL / 4; ++kt) {
            const int k = kt * 4;
            v2f a = *(const v2f*)(Xs + mrow * XSTRIDE + k + khalf);
            if (PACKED) {
                // tile index = (e*128 + kt)*128 + nt_global ; 64 floats per tile
                const v2f* bp = (const v2f*)(pk1 +
                    (((size_t)(e * (D_MODEL / 4) + kt)) * (HIDDEN / 16) + (colbase >> 4)) * 64) + lane;
#pragma unroll
                for (int nt = 0; nt < 4; ++nt) {
                    v2f b = bp[nt * 32];   // +64 floats per N-tile
                    hacc[nt] = __builtin_amdgcn_wmma_f32_16x16x4_f32(
                        false, a, false, b, (short)0, hacc[nt], false, false);
                }
            } else {
                const float* bp = w1e + (size_t)(k + khalf) * HIDDEN + colbase + ncol;
#pragma unroll
                for (int nt = 0; nt < 4; ++nt) {
                    v2f b;
                    b.x = bp[nt * 16];
                    b.y = bp[nt * 16 + HIDDEN];
                    hacc[nt] = __builtin_amdgcn_wmma_f32_16x16x4_f32(
                        false, a, false, b, (short)0, hacc[nt], false, false);
                }
            }
        }

        // bias + exact gelu, write chunk to LDS (C-layout -> A-layout transpose)
#pragma unroll
        for (int nt = 0; nt < 4; ++nt) {
            const int col  = colbase + nt * 16 + ncol;
            const float bb = b1[e * HIDDEN + col];
            const int lcol = q * 64 + nt * 16 + ncol;
#pragma unroll
            for (int j = 0; j < 8; ++j) {
                float h = hacc[nt][j] + bb;
                float g = 0.5f * h * (1.f + erff(h * 0.70710678118654752f));
                int row = mg * 16 + j + ((lane >= 16) ? 8 : 0);
                Hs[row * HSTRIDE + lcol] = g;
            }
        }
        __syncthreads();

        // ---- GEMM2: Y[16 x 128] per wave += H[16x256] @ W2[rows, cols] ----
        for (int kt = 0; kt < HCHUNK / 4; ++kt) {
            const int k = kt * 4;
            v2f a = *(const v2f*)(Hs + mrow * HSTRIDE + k + khalf);
            if (PACKED) {
                const v2f* bp = (const v2f*)(pk2 +
                    (((size_t)(e * (HIDDEN / 4) + hc * (HCHUNK / 4) + kt)) * (D_MODEL / 16) + q * 8) * 64) + lane;
#pragma unroll
                for (int nt = 0; nt < 8; ++nt) {
                    v2f b = bp[nt * 32];
                    yacc[nt] = __builtin_amdgcn_wmma_f32_16x16x4_f32(
                        false, a, false, b, (short)0, yacc[nt], false, false);
                }
            } else {
                const float* bp = w2e + (size_t)(hc * HCHUNK + k + khalf) * D_MODEL + q * 128 + ncol;
#pragma unroll
                for (int nt = 0; nt < 8; ++nt) {
                    v2f b;
                    b.x = bp[nt * 16];
                    b.y = bp[nt * 16 + D_MODEL];
                    yacc[nt] = __builtin_amdgcn_wmma_f32_16x16x4_f32(
                        false, a, false, b, (short)0, yacc[nt], false, false);
                }
            }
        }
        __syncthreads();   // protect Hs before next chunk overwrites it
    }

    // ---- epilogue: (Y + b2) * routing_weight, scatter-add ----
#pragma unroll
    for (int nt = 0; nt < 8; ++nt) {
        const int col  = q * 128 + nt * 16 + ncol;
        const float bb = b2[e * D_MODEL + col];
#pragma unroll
        for (int j = 0; j < 8; ++j) {
            int row = mg * 16 + j + ((lane >= 16) ? 8 : 0);
            float val = (yacc[nt][j] + bb) * wgt[row];
            atomicAdd(out + (size_t)toks[row] * D_MODEL + col, val);
        }
    }
}

// ---------------- launch ----------------
extern "C" void kernel_launch(void* const* d_in, const int* in_sizes, int n_in,
                              void* d_out, int out_size, void* d_ws, size_t ws_size,
                              hipStream_t stream) {
    (void)in_sizes; (void)n_in; (void)out_size;
    const float* x  = (const float*)d_in[0];
    const float* ts = (const float*)d_in[1];
    const float* gw = (const float*)d_in[2];
    const float* gb = (const float*)d_in[3];
    const float* w1 = (const float*)d_in[4];
    const float* b1 = (const float*)d_in[5];
    const float* w2 = (const float*)d_in[6];
    const float* b2 = (const float*)d_in[7];
    float* out = (float*)d_out;

    int*   cnt = (int*)d_ws;
    int*   idx = (int*)((char*)d_ws + 256);
    float* wg  = (float*)((char*)d_ws + 256 + (size_t)NEXP * TOKENS * sizeof(int));

    const size_t PK_BYTES = (size_t)NEXP * D_MODEL * HIDDEN * sizeof(float); // 33.55MB each
    const size_t PK1_OFF  = (size_t)1 << 20;
    const size_t PK2_OFF  = PK1_OFF + PK_BYTES;
    const bool packed     = ws_size >= PK2_OFF + PK_BYTES;
    float* pk1 = (float*)((char*)d_ws + PK1_OFF);
    float* pk2 = (float*)((char*)d_ws + PK2_OFF);

    moe_zero<<<1024, 256, 0, stream>>>(out, cnt, TOKENS * D_MODEL / 4);
    moe_router<<<TOKENS / 8, 256, 0, stream>>>(x, ts, gw, gb, cnt, idx, wg);

    dim3 grid(MAX_TILES, NEXP);
    if (packed) {
        // W1: E*D rows of width HIDDEN ; W2: E*HIDDEN rows of width D
        moe_pack<<<2048, 256, 0, stream>>>(w1, pk1, NEXP * D_MODEL / 4, HIDDEN / 16);
        moe_pack<<<2048, 256, 0, stream>>>(w2, pk2, NEXP * HIDDEN / 4, D_MODEL / 16);
        moe_expert<true><<<grid, 256, 0, stream>>>(x, w1, b1, w2, b2, pk1, pk2,
                                                   cnt, idx, wg, out);
    } else {
        moe_expert<false><<<grid, 256, 0, stream>>>(x, w1, b1, w2, b2, pk1, pk2,
                                                    cnt, idx, wg, out);
    }
}